// GCNLayer_84902913507383
// MI455X (gfx1250) — compile-verified
//
#include <hip/hip_runtime.h>

typedef __attribute__((ext_vector_type(2))) float v2f;
typedef __attribute__((ext_vector_type(8))) float v8f;

#define N_NODES 100000
#define N_EDGES 1600000
#define E2      (2 * N_EDGES)
#define IN_CH   512
#define HID     16
#define OUT_CH  64
#define NTILES  (N_NODES / 16)   // 6250, exact

// ---------------------------------------------------------------------------
// deg[r] += w  over the concatenated edge list (rows only, per reference)
// ---------------------------------------------------------------------------
__global__ void __launch_bounds__(256) deg_kernel(
    const int* __restrict__ ei1, const float* __restrict__ ew1,
    const int* __restrict__ ei2, const float* __restrict__ ew2,
    float* __restrict__ deg) {
  int e = blockIdx.x * blockDim.x + threadIdx.x;
  if (e >= E2) return;
  int r; float w;
  if (e < N_EDGES) { r = ei1[e]; w = ew1[e]; }
  else             { int t = e - N_EDGES; r = ei2[t]; w = ew2[t]; }
  atomicAdd(&deg[r], w);
}

// deg -> dinv in place
__global__ void __launch_bounds__(256) dinv_kernel(float* __restrict__ deg) {
  int i = blockIdx.x * blockDim.x + threadIdx.x;
  if (i >= N_NODES) return;
  float d = deg[i];
  deg[i] = (d > 0.0f) ? rsqrtf(fmaxf(d, 1e-12f)) : 0.0f;
}

// norm[e] = dinv[row]*w*dinv[col]
__global__ void __launch_bounds__(256) norm_kernel(
    const int* __restrict__ ei1, const float* __restrict__ ew1,
    const int* __restrict__ ei2, const float* __restrict__ ew2,
    const float* __restrict__ dinv, float* __restrict__ nrm) {
  int e = blockIdx.x * blockDim.x + threadIdx.x;
  if (e >= E2) return;
  int r, c; float w;
  if (e < N_EDGES) { r = ei1[e]; c = ei1[N_EDGES + e]; w = ew1[e]; }
  else             { int t = e - N_EDGES; r = ei2[t]; c = ei2[N_EDGES + t]; w = ew2[t]; }
  nrm[e] = dinv[r] * w * dinv[c];
}

// ---------------------------------------------------------------------------
// GEMM1: h1[100000x16] = x[100000x512] @ W1[512x16]  via V_WMMA_F32_16X16X4_F32
// 8 waves/block, one 16-row tile per wave, K looped 4 at a time (128 WMMAs).
// W1 staged in LDS as (k-pair, n) float2: lane<16 reads K=k..k+1, lane>=16
// reads K=k+2..k+3 -> matches documented f32 A/B VGPR lane layout.
// ---------------------------------------------------------------------------
__global__ void __launch_bounds__(256) gemm1_kernel(
    const float* __restrict__ x, const float* __restrict__ W1,
    float* __restrict__ h1) {
  __shared__ v2f ldsB[(IN_CH / 2) * HID];  // 4096 * 8B = 32 KB
  for (int i = threadIdx.x; i < (IN_CH / 2) * HID; i += 256) {
    int kp = i >> 4, n = i & 15;
    v2f t;
    t.x = W1[(2 * kp) * HID + n];
    t.y = W1[(2 * kp + 1) * HID + n];
    ldsB[i] = t;
  }
  __syncthreads();

  int lane = threadIdx.x & 31;
  int wv   = threadIdx.x >> 5;
  int tile = blockIdx.x * 8 + wv;          // wave-uniform guard -> EXEC all-1s
  if (tile < NTILES) {
    int lo = lane & 15, hi = lane >> 4;
    const float* ap = x + (size_t)(tile * 16 + lo) * IN_CH + 2 * hi;
    v8f acc = {};
#pragma unroll 4
    for (int k = 0; k < IN_CH; k += 4) {
      v2f a = *reinterpret_cast<const v2f*>(ap + k);          // A: rows on lanes
      v2f b = ldsB[((k >> 1) + hi) * HID + lo];               // B: cols on lanes
      acc = __builtin_amdgcn_wmma_f32_16x16x4_f32(
          false, a, false, b, (short)0, acc, false, false);
    }
    float* out = h1 + (size_t)(tile * 16) * HID + lo;
#pragma unroll
    for (int r = 0; r < 8; ++r)
      out[(size_t)(r + 8 * hi) * HID] = acc[r];
  }
}

// ---------------------------------------------------------------------------
// scatter16: agg1[row] += norm[e] * h1[col]   (one lane per (edge, channel))
// ---------------------------------------------------------------------------
__global__ void __launch_bounds__(256) scatter16_kernel(
    const int* __restrict__ ei1, const int* __restrict__ ei2,
    const float* __restrict__ nrm, const float* __restrict__ h1,
    float* __restrict__ agg1) {
  int idx = blockIdx.x * blockDim.x + threadIdx.x;   // exact grid: E2*16
  int e = idx >> 4, c = idx & 15;
  int r, co;
  if (e < N_EDGES) { r = ei1[e]; co = ei1[N_EDGES + e]; }
  else             { int t = e - N_EDGES; r = ei2[t]; co = ei2[N_EDGES + t]; }
  atomicAdd(&agg1[(size_t)r * HID + c], nrm[e] * h1[(size_t)co * HID + c]);
}

// ---------------------------------------------------------------------------
// GEMM2: h2[100000x64] = relu(agg1 + b1) @ W2[16x64], bias+ReLU fused into A.
// Per wave: one 16-row tile, 4 N-tiles of 16, K=16 in 4 WMMA steps.
// ---------------------------------------------------------------------------
__global__ void __launch_bounds__(256) gemm2_kernel(
    const float* __restrict__ agg1, const float* __restrict__ b1,
    const float* __restrict__ W2, float* __restrict__ h2) {
  __shared__ v2f   ldsB[(HID / 2) * OUT_CH];  // 512 * 8B = 4 KB
  __shared__ float ldsb1[HID];
  if (threadIdx.x < HID) ldsb1[threadIdx.x] = b1[threadIdx.x];
  for (int i = threadIdx.x; i < (HID / 2) * OUT_CH; i += 256) {
    int kp = i >> 6, n = i & 63;
    v2f t;
    t.x = W2[(2 * kp) * OUT_CH + n];
    t.y = W2[(2 * kp + 1) * OUT_CH + n];
    ldsB[i] = t;
  }
  __syncthreads();

  int lane = threadIdx.x & 31;
  int wv   = threadIdx.x >> 5;
  int tile = blockIdx.x * 8 + wv;
  if (tile < NTILES) {
    int lo = lane & 15, hi = lane >> 4;
    const float* ap = agg1 + (size_t)(tile * 16 + lo) * HID + 2 * hi;
    v8f a0 = {}, a1 = {}, a2 = {}, a3 = {};
#pragma unroll
    for (int k = 0; k < HID; k += 4) {
      v2f a;
      a.x = fmaxf(ap[k]     + ldsb1[k + 2 * hi],     0.0f);  // fused bias+ReLU
      a.y = fmaxf(ap[k + 1] + ldsb1[k + 2 * hi + 1], 0.0f);
      const v2f* brow = &ldsB[((k >> 1) + hi) * OUT_CH];
      a0 = __builtin_amdgcn_wmma_f32_16x16x4_f32(false, a, false, brow[lo],      (short)0, a0, false, false);
      a1 = __builtin_amdgcn_wmma_f32_16x16x4_f32(false, a, false, brow[16 + lo], (short)0, a1, false, false);
      a2 = __builtin_amdgcn_wmma_f32_16x16x4_f32(false, a, false, brow[32 + lo], (short)0, a2, false, false);
      a3 = __builtin_amdgcn_wmma_f32_16x16x4_f32(false, a, false, brow[48 + lo], (short)0, a3, false, false);
    }
    float* out = h2 + (size_t)(tile * 16) * OUT_CH;
#pragma unroll
    for (int r = 0; r < 8; ++r) {
      int row = r + 8 * hi;
      out[(size_t)row * OUT_CH + lo]      = a0[r];
      out[(size_t)row * OUT_CH + 16 + lo] = a1[r];
      out[(size_t)row * OUT_CH + 32 + lo] = a2[r];
      out[(size_t)row * OUT_CH + 48 + lo] = a3[r];
    }
  }
}

// ---------------------------------------------------------------------------
// scatter64: out[row] += norm[e] * h2[col]   (accumulates directly into d_out)
// ---------------------------------------------------------------------------
__global__ void __launch_bounds__(256) scatter64_kernel(
    const int* __restrict__ ei1, const int* __restrict__ ei2,
    const float* __restrict__ nrm, const float* __restrict__ h2,
    float* __restrict__ out) {
  int idx = blockIdx.x * blockDim.x + threadIdx.x;   // exact grid: E2*64
  int e = idx >> 6, c = idx & 63;
  int r, co;
  if (e < N_EDGES) { r = ei1[e]; co = ei1[N_EDGES + e]; }
  else             { int t = e - N_EDGES; r = ei2[t]; co = ei2[N_EDGES + t]; }
  atomicAdd(&out[(size_t)r * OUT_CH + c], nrm[e] * h2[(size_t)co * OUT_CH + c]);
}

// ---------------------------------------------------------------------------
// log_softmax over 64 channels, in place on d_out (+b2). One wave per row,
// 2 channels per lane, wave32 shfl_xor reductions.
// ---------------------------------------------------------------------------
__global__ void __launch_bounds__(256) logsoftmax_kernel(
    float* __restrict__ out, const float* __restrict__ b2) {
  int lane = threadIdx.x & 31;
  int wv   = threadIdx.x >> 5;
  int row  = blockIdx.x * 8 + wv;                    // exact grid: N_NODES/8
  float* p = out + (size_t)row * OUT_CH;
  float v0 = p[lane]      + b2[lane];
  float v1 = p[32 + lane] + b2[32 + lane];
  float m = fmaxf(v0, v1);
#pragma unroll
  for (int off = 16; off >= 1; off >>= 1)
    m = fmaxf(m, __shfl_xor(m, off, 32));
  float s = __expf(v0 - m) + __expf(v1 - m);
#pragma unroll
  for (int off = 16; off >= 1; off >>= 1)
    s += __shfl_xor(s, off, 32);
  float ls = __logf(s);
  p[lane]      = v0 - m - ls;
  p[32 + lane] = v1 - m - ls;
}

// ---------------------------------------------------------------------------
extern "C" void kernel_launch(void* const* d_in, const int* in_sizes, int n_in,
                              void* d_out, int out_size, void* d_ws, size_t ws_size,
                              hipStream_t stream) {
  (void)in_sizes; (void)n_in; (void)out_size; (void)ws_size;
  const float* x   = (const float*)d_in[0];
  const int*   ei1 = (const int*)d_in[1];
  const float* ew1 = (const float*)d_in[2];
  const int*   ei2 = (const int*)d_in[3];
  const float* ew2 = (const float*)d_in[4];
  const float* W1  = (const float*)d_in[5];
  const float* b1  = (const float*)d_in[6];
  const float* W2  = (const float*)d_in[7];
  const float* b2  = (const float*)d_in[8];

  float* ws   = (float*)d_ws;
  float* deg  = ws;             // 100000 floats (becomes dinv in place)
  float* nrm  = ws + 100032;    // 3200000 floats
  float* h1   = ws + 3300032;   // 1600000 floats
  float* agg1 = ws + 4900032;   // 1600000 floats
  float* h2   = ws + 6500032;   // 6400000 floats  (total ~51.6 MB)
  float* out  = (float*)d_out;  // layer-2 aggregation accumulates here

  hipMemsetAsync(deg,  0, (size_t)N_NODES * sizeof(float), stream);
  hipMemsetAsync(agg1, 0, (size_t)N_NODES * HID * sizeof(float), stream);
  hipMemsetAsync(out,  0, (size_t)N_NODES * OUT_CH * sizeof(float), stream);

  deg_kernel       <<<E2 / 256, 256, 0, stream>>>(ei1, ew1, ei2, ew2, deg);
  dinv_kernel      <<<(N_NODES + 255) / 256, 256, 0, stream>>>(deg);
  norm_kernel      <<<E2 / 256, 256, 0, stream>>>(ei1, ew1, ei2, ew2, deg, nrm);
  gemm1_kernel     <<<(NTILES + 7) / 8, 256, 0, stream>>>(x, W1, h1);
  scatter16_kernel <<<(E2 * HID) / 256, 256, 0, stream>>>(ei1, ei2, nrm, h1, agg1);
  gemm2_kernel     <<<(NTILES + 7) / 8, 256, 0, stream>>>(agg1, b1, W2, h2);
  scatter64_kernel <<<(E2 * OUT_CH) / 256, 256, 0, stream>>>(ei1, ei2, nrm, h2, out);
  logsoftmax_kernel<<<N_NODES / 8, 256, 0, stream>>>(out, b2);
}